// MeanReadout_18880676233586
// MI455X (gfx1250) — compile-verified
//
#include <hip/hip_runtime.h>

typedef float v2f __attribute__((ext_vector_type(2)));
typedef float v8f __attribute__((ext_vector_type(8)));

#define DFEAT          256
#define THREADS        512        // 16 wave32s -> 16 stripes x 16 cols = 256 features
#define ROWS_PER_BLOCK 1024
#define SENT           ((int)0x80000000)

__global__ void zero_kernel(float* __restrict__ out, int n) {
  int i = blockIdx.x * blockDim.x + threadIdx.x;
  if (i < n) out[i] = 0.0f;
}

// Segmented sum via V_WMMA_F32_16X16X4_F32:
//   D(16x16,f32) += A(16x4 selection) x B(4x16 node tile)
// Row m of D is the running f32 sum of rows assigned to graph-slot m.
// Sorted ids -> fast path: a 16-row group is segment-uniform iff first==last.
__global__ __launch_bounds__(THREADS)
void segsum_wmma_kernel(const float* __restrict__ nodes,
                        const int*   __restrict__ gid,
                        float*       __restrict__ sums,
                        int nNodes) {
  __shared__ alignas(16) int sgid[ROWS_PER_BLOCK];

  const int rowStart = blockIdx.x * ROWS_PER_BLOCK;
  const int rowEnd   = min(rowStart + ROWS_PER_BLOCK, nNodes);
  const int rowsHere = rowEnd - rowStart;

  for (int i = threadIdx.x; i < rowsHere; i += THREADS)
    sgid[i] = gid[rowStart + i];
  __syncthreads();

  const int tid     = threadIdx.x;
  const int wave    = tid >> 5;          // 0..15
  const int lane    = tid & 31;
  const int laneHi  = lane >> 4;         // 0: lanes 0-15, 1: lanes 16-31
  const int laneN   = lane & 15;         // column within stripe / M index for A
  const int colBase = wave * 16;
  const int rowOff  = laneHi ? 2 : 0;    // B rows K2/K3 live in lanes 16-31

  const v8f zero8 = {};
  v8f c = zero8;                         // 16x16 f32 accumulator (slots x cols)
  int myG[8];                            // graph id of slot (v + laneHi*8)
  int curSlot = -1;
  int curGid  = SENT;

  auto flushAll = [&]() {
#pragma unroll
    for (int v = 0; v < 8; ++v) {
      int slotIdx = v + (laneHi << 3);
      if (slotIdx <= curSlot)
        atomicAdd(&sums[(size_t)myG[v] * DFEAT + colBase + laneN], c[v]);
    }
    c = zero8;
    curSlot = -1;
    curGid  = SENT;
  };

  auto openIfNew = [&](int g) {
    if (__builtin_amdgcn_readfirstlane((int)(g != curGid))) {
      if (__builtin_amdgcn_readfirstlane((int)(curSlot >= 15))) flushAll();
      curSlot++;
      curGid = g;
      int half = curSlot >> 3, tv = curSlot & 7;
      bool mine = (laneHi == half);
#pragma unroll
      for (int v = 0; v < 8; ++v) if (mine && tv == v) myG[v] = g;
    }
  };

  for (int k0 = rowStart; k0 < rowEnd; k0 += 16) {
    const int li      = k0 - rowStart;
    const bool full16 = (k0 + 16) <= rowEnd;
    const int  gF     = sgid[li];

    // ---- fast path: 16 segment-uniform rows (sorted => first==last) ----
    if (full16 && __builtin_amdgcn_readfirstlane((int)(gF == sgid[li + 15]))) {
      openIfNew(gF);
      const float av = (curSlot == laneN) ? 1.0f : 0.0f;
      v2f a; a.x = av; a.y = av;       // A[m][k] identical for all K
      const float* p = nodes + (size_t)(k0 + rowOff) * DFEAT + colBase + laneN;
#pragma unroll
      for (int j = 0; j < 4; ++j) {
        v2f b;
        b.x = __builtin_nontemporal_load(p + (size_t)(4 * j) * DFEAT);
        b.y = __builtin_nontemporal_load(p + (size_t)(4 * j) * DFEAT + DFEAT);
        c = __builtin_amdgcn_wmma_f32_16x16x4_f32(false, a, false, b,
                                                  (short)0, c, false, false);
      }
      continue;
    }

    // ---- boundary / tail group: 4-row sub-groups ----
    for (int k4 = k0; k4 < k0 + 16 && k4 < rowEnd; k4 += 4) {
      const int l4 = k4 - rowStart;
      int4 gg = *reinterpret_cast<const int4*>(&sgid[l4]);
      const bool full4 = (k4 + 4) <= rowEnd;

      // sub-fast: 4 segment-uniform rows
      if (full4 && __builtin_amdgcn_readfirstlane((int)(gg.x == gg.w))) {
        openIfNew(gg.x);
        const float av = (curSlot == laneN) ? 1.0f : 0.0f;
        v2f a; a.x = av; a.y = av;
        const float* p = nodes + (size_t)(k4 + rowOff) * DFEAT + colBase + laneN;
        v2f b;
        b.x = __builtin_nontemporal_load(p);
        b.y = __builtin_nontemporal_load(p + DFEAT);
        c = __builtin_amdgcn_wmma_f32_16x16x4_f32(false, a, false, b,
                                                  (short)0, c, false, false);
        continue;
      }

      // general: rows may open several new slots (rare: one per segment)
      const bool vld1 = (k4 + 1) < rowEnd;
      const bool vld2 = (k4 + 2) < rowEnd;
      const bool vld3 = (k4 + 3) < rowEnd;

      int t = 0, prev = curGid;
      if (gg.x != prev)         { t++; prev = gg.x; }
      if (vld1 && gg.y != prev) { t++; prev = gg.y; }
      if (vld2 && gg.z != prev) { t++; prev = gg.z; }
      if (vld3 && gg.w != prev) { t++; prev = gg.w; }
      if (__builtin_amdgcn_readfirstlane((int)(curSlot + t > 15))) flushAll();

      int s0, s1, s2, s3;
      {
        int g0 = gg.x;
        if (g0 != curGid) {
          curSlot++; curGid = g0;
          int half = curSlot >> 3, tv = curSlot & 7; bool mine = (laneHi == half);
#pragma unroll
          for (int v = 0; v < 8; ++v) if (mine && tv == v) myG[v] = g0;
        }
        s0 = curSlot;
      }
      {
        int g1 = gg.y;
        if (vld1 && g1 != curGid) {
          curSlot++; curGid = g1;
          int half = curSlot >> 3, tv = curSlot & 7; bool mine = (laneHi == half);
#pragma unroll
          for (int v = 0; v < 8; ++v) if (mine && tv == v) myG[v] = g1;
        }
        s1 = vld1 ? curSlot : -1;
      }
      {
        int g2 = gg.z;
        if (vld2 && g2 != curGid) {
          curSlot++; curGid = g2;
          int half = curSlot >> 3, tv = curSlot & 7; bool mine = (laneHi == half);
#pragma unroll
          for (int v = 0; v < 8; ++v) if (mine && tv == v) myG[v] = g2;
        }
        s2 = vld2 ? curSlot : -1;
      }
      {
        int g3 = gg.w;
        if (vld3 && g3 != curGid) {
          curSlot++; curGid = g3;
          int half = curSlot >> 3, tv = curSlot & 7; bool mine = (laneHi == half);
#pragma unroll
          for (int v = 0; v < 8; ++v) if (mine && tv == v) myG[v] = g3;
        }
        s3 = vld3 ? curSlot : -1;
      }

      v2f a;
      if (laneHi == 0) {
        a.x = (s0 == laneN) ? 1.0f : 0.0f;
        a.y = (s1 == laneN) ? 1.0f : 0.0f;
      } else {
        a.x = (s2 == laneN) ? 1.0f : 0.0f;
        a.y = (s3 == laneN) ? 1.0f : 0.0f;
      }

      int r0 = k4 + rowOff;
      int r1 = r0 + 1;
      r0 = min(r0, nNodes - 1);          // clamp tail (A row is zeroed anyway)
      r1 = min(r1, nNodes - 1);
      v2f b;
      b.x = __builtin_nontemporal_load(&nodes[(size_t)r0 * DFEAT + colBase + laneN]);
      b.y = __builtin_nontemporal_load(&nodes[(size_t)r1 * DFEAT + colBase + laneN]);

      c = __builtin_amdgcn_wmma_f32_16x16x4_f32(false, a, false, b,
                                                (short)0, c, false, false);
    }
  }

  flushAll();                             // final flush of live slots
}

__device__ __forceinline__ int lower_bound_dev(const int* __restrict__ a, int n, int val) {
  int lo = 0, hi = n;
  while (lo < hi) {
    int mid = (lo + hi) >> 1;
    if (a[mid] < val) lo = mid + 1; else hi = mid;
  }
  return lo;
}

// In-place mean: counts recovered from the sorted graph_id via binary search.
__global__ void divide_kernel(float* __restrict__ out,
                              const int* __restrict__ gid,
                              int nNodes) {
  __shared__ int scnt;
  const int g = blockIdx.x;
  if (threadIdx.x == 0) {
    int lo = lower_bound_dev(gid, nNodes, g);
    int hi = lower_bound_dev(gid, nNodes, g + 1);
    scnt = hi - lo;
  }
  __syncthreads();
  const float d = (float)scnt;           // 0/0 -> NaN, matching the reference
  size_t idx = (size_t)g * DFEAT + threadIdx.x;
  out[idx] = out[idx] / d;
}

extern "C" void kernel_launch(void* const* d_in, const int* in_sizes, int n_in,
                              void* d_out, int out_size, void* d_ws, size_t ws_size,
                              hipStream_t stream) {
  const float* nodes = (const float*)d_in[0];
  const int*   gid   = (const int*)d_in[1];     // sorted graph ids
  (void)d_ws; (void)ws_size; (void)n_in;

  const int nNodes    = in_sizes[1];            // 1,000,000
  const int numGraphs = out_size / DFEAT;       // 8192
  float* out = (float*)d_out;

  // 1) zero the accumulator (d_out is poisoned / carries prior results)
  zero_kernel<<<(out_size + 255) / 256, 256, 0, stream>>>(out, out_size);

  // 2) WMMA segmented sum with boundary atomics
  int nBlocks = (nNodes + ROWS_PER_BLOCK - 1) / ROWS_PER_BLOCK;
  segsum_wmma_kernel<<<nBlocks, THREADS, 0, stream>>>(nodes, gid, out, nNodes);

  // 3) in-place divide by segment counts (binary search on sorted ids)
  divide_kernel<<<numGraphs, DFEAT, 0, stream>>>(out, gid, nNodes);
}